// GATNet_7198365188474
// MI455X (gfx1250) — compile-verified
//
#include <hip/hip_runtime.h>
#include <math.h>

typedef __attribute__((ext_vector_type(2))) float v2f;
typedef __attribute__((ext_vector_type(8))) float v8f;

#define N_NODES 50000
#define N_EDGES 800000
#define EP (N_EDGES + N_NODES)   // edges + self loops
#define IN_C 256
#define HID 64
#define HEADS 8
#define F1 (HEADS * HID)         // 512
#define OUT_C 64

// ---------------------------------------------------------------------------
// helpers
// ---------------------------------------------------------------------------
__device__ __forceinline__ void edge_sd(const int* __restrict__ ei, int e, int& s, int& d) {
    if (e < N_EDGES) { s = ei[e]; d = ei[N_EDGES + e]; }
    else             { s = e - N_EDGES; d = s; }
}

__device__ __forceinline__ void atomicMaxFloat(float* addr, float val) {
    // standard monotone-punning trick: works for mixed signs, init must be -inf
    if (val >= 0.0f) atomicMax((int*)addr, __float_as_int(val));
    else             atomicMin((unsigned int*)addr, __float_as_uint(val));
}

__device__ __forceinline__ float leaky02(float v) { return v > 0.0f ? v : 0.2f * v; }

// ---------------------------------------------------------------------------
// WMMA f32 GEMM:  C[M,NCOL] = A[M,K] @ B[K,NCOL]   (M,K,NCOL all multiples of 16/4)
// one wave -> one 16x16 output tile, K stepped by 4 (V_WMMA_F32_16X16X4_F32)
// ---------------------------------------------------------------------------
__global__ void gemm_wmma_f32(const float* __restrict__ A, const float* __restrict__ B,
                              float* __restrict__ C, int M, int K, int NCOL) {
    const int lane = threadIdx.x & 31;
    const int wave = threadIdx.x >> 5;
    const int ntn  = NCOL >> 4;
    const int tile = blockIdx.x * (blockDim.x >> 5) + wave;
    if (tile >= (M >> 4) * ntn) return;          // wave-uniform: EXEC stays all-1s
    const int m0 = (tile / ntn) << 4;
    const int n0 = (tile % ntn) << 4;

    const int mr = m0 + (lane & 15);             // A row for this lane
    const int kh = (lane >> 4) << 1;             // lanes 0-15 -> K+{0,1}, 16-31 -> K+{2,3}
    const int nc = n0 + (lane & 15);             // B/C column for this lane

    v8f acc = {};
    #pragma unroll 4
    for (int k = 0; k < K; k += 4) {
        const float* ap = A + (size_t)mr * K + (k + kh);
        v2f a; a.x = ap[0]; a.y = ap[1];
        const float* bp = B + (size_t)(k + kh) * NCOL + nc;
        v2f b; b.x = bp[0]; b.y = bp[NCOL];
        acc = __builtin_amdgcn_wmma_f32_16x16x4_f32(false, a, false, b,
                                                    (short)0, acc, false, false);
    }

    const int crow = m0 + (lane >> 4) * 8;
    float* cp = C + (size_t)crow * NCOL + nc;
    #pragma unroll
    for (int r = 0; r < 8; ++r) cp[(size_t)r * NCOL] = acc[r];
}

// ---------------------------------------------------------------------------
// attention coefficients, layer 1: one wave per node, 8 heads x 64 channels
// ---------------------------------------------------------------------------
__global__ void alpha1_kernel(const float* __restrict__ h1,
                              const float* __restrict__ a_src, const float* __restrict__ a_dst,
                              float* __restrict__ as, float* __restrict__ ad) {
    const int node = blockIdx.x * (blockDim.x >> 5) + (threadIdx.x >> 5);
    if (node >= N_NODES) return;
    const int lane = threadIdx.x & 31;
    const float* hp = h1 + (size_t)node * F1;
    #pragma unroll
    for (int h = 0; h < HEADS; ++h) {
        const float v0 = hp[h * HID + lane];
        const float v1 = hp[h * HID + lane + 32];
        float s = v0 * a_src[h * HID + lane] + v1 * a_src[h * HID + lane + 32];
        float d = v0 * a_dst[h * HID + lane] + v1 * a_dst[h * HID + lane + 32];
        #pragma unroll
        for (int off = 16; off > 0; off >>= 1) {
            s += __shfl_down(s, off, 32);
            d += __shfl_down(d, off, 32);
        }
        if (lane == 0) { as[node * HEADS + h] = s; ad[node * HEADS + h] = d; }
    }
}

// ---------------------------------------------------------------------------
// init accumulators for both layers
// ---------------------------------------------------------------------------
__global__ void init_kernel(float* __restrict__ out1, float* __restrict__ emax1,
                            float* __restrict__ esum1, float* __restrict__ out2,
                            float* __restrict__ emax2, float* __restrict__ esum2) {
    const size_t i = (size_t)blockIdx.x * blockDim.x + threadIdx.x;
    if (i < (size_t)N_NODES * F1) out1[i] = 0.0f;
    if (i < (size_t)N_NODES * HEADS) { emax1[i] = -INFINITY; esum1[i] = 0.0f; }
    if (i < (size_t)N_NODES * OUT_C) out2[i] = 0.0f;
    if (i < (size_t)N_NODES) { emax2[i] = -INFINITY; esum2[i] = 0.0f; }
}

// ---------------------------------------------------------------------------
// segment max / sum-exp passes, layer 1 (thread per edge-head)
// ---------------------------------------------------------------------------
__global__ void max1_kernel(const int* __restrict__ ei, const float* __restrict__ as,
                            const float* __restrict__ ad, float* __restrict__ emax) {
    const int idx = blockIdx.x * blockDim.x + threadIdx.x;
    if (idx >= EP * HEADS) return;
    const int e = idx >> 3, h = idx & 7;
    int s, d; edge_sd(ei, e, s, d);
    const float v = leaky02(as[s * HEADS + h] + ad[d * HEADS + h]);
    atomicMaxFloat(&emax[d * HEADS + h], v);
}

__global__ void sum1_kernel(const int* __restrict__ ei, const float* __restrict__ as,
                            const float* __restrict__ ad, const float* __restrict__ emax,
                            float* __restrict__ esum) {
    const int idx = blockIdx.x * blockDim.x + threadIdx.x;
    if (idx >= EP * HEADS) return;
    const int e = idx >> 3, h = idx & 7;
    int s, d; edge_sd(ei, e, s, d);
    const float v = leaky02(as[s * HEADS + h] + ad[d * HEADS + h]);
    atomicAdd(&esum[d * HEADS + h], __expf(v - emax[d * HEADS + h]));
}

// ---------------------------------------------------------------------------
// message aggregation, layer 1: one wave per edge, coalesced over 512 channels
// ---------------------------------------------------------------------------
__global__ void agg1_kernel(const int* __restrict__ ei, const float* __restrict__ h1,
                            const float* __restrict__ as, const float* __restrict__ ad,
                            const float* __restrict__ emax, const float* __restrict__ esum,
                            float* __restrict__ out1) {
    const int e = blockIdx.x * (blockDim.x >> 5) + (threadIdx.x >> 5);
    if (e >= EP) return;
    const int lane = threadIdx.x & 31;
    int s, d; edge_sd(ei, e, s, d);
    float al = 0.0f;
    if (lane < HEADS) {
        const float v = leaky02(as[s * HEADS + lane] + ad[d * HEADS + lane]);
        al = __expf(v - emax[d * HEADS + lane]) / esum[d * HEADS + lane];
    }
    const float* hp = h1 + (size_t)s * F1;
    float* op = out1 + (size_t)d * F1;
    #pragma unroll
    for (int h = 0; h < HEADS; ++h) {
        const float a = __shfl(al, h, 32);
        atomicAdd(&op[h * HID + lane],      hp[h * HID + lane]      * a);
        atomicAdd(&op[h * HID + lane + 32], hp[h * HID + lane + 32] * a);
    }
}

// ---------------------------------------------------------------------------
// bias + ELU (in place): g = elu(out1 + b1[col])
// ---------------------------------------------------------------------------
__global__ void post1_kernel(float* __restrict__ g, const float* __restrict__ b1) {
    const size_t i = (size_t)blockIdx.x * blockDim.x + threadIdx.x;
    if (i >= (size_t)N_NODES * F1) return;
    const float v = g[i] + b1[i % F1];
    g[i] = v > 0.0f ? v : (__expf(v) - 1.0f);
}

// ---------------------------------------------------------------------------
// layer 2: single head, C = 64
// ---------------------------------------------------------------------------
__global__ void alpha2_kernel(const float* __restrict__ h2, const float* __restrict__ a_src,
                              const float* __restrict__ a_dst,
                              float* __restrict__ as, float* __restrict__ ad) {
    const int node = blockIdx.x * (blockDim.x >> 5) + (threadIdx.x >> 5);
    if (node >= N_NODES) return;
    const int lane = threadIdx.x & 31;
    const float* hp = h2 + (size_t)node * OUT_C;
    const float v0 = hp[lane], v1 = hp[lane + 32];
    float s = v0 * a_src[lane] + v1 * a_src[lane + 32];
    float d = v0 * a_dst[lane] + v1 * a_dst[lane + 32];
    #pragma unroll
    for (int off = 16; off > 0; off >>= 1) {
        s += __shfl_down(s, off, 32);
        d += __shfl_down(d, off, 32);
    }
    if (lane == 0) { as[node] = s; ad[node] = d; }
}

__global__ void max2_kernel(const int* __restrict__ ei, const float* __restrict__ as,
                            const float* __restrict__ ad, float* __restrict__ emax) {
    const int e = blockIdx.x * blockDim.x + threadIdx.x;
    if (e >= EP) return;
    int s, d; edge_sd(ei, e, s, d);
    atomicMaxFloat(&emax[d], leaky02(as[s] + ad[d]));
}

__global__ void sum2_kernel(const int* __restrict__ ei, const float* __restrict__ as,
                            const float* __restrict__ ad, const float* __restrict__ emax,
                            float* __restrict__ esum) {
    const int e = blockIdx.x * blockDim.x + threadIdx.x;
    if (e >= EP) return;
    int s, d; edge_sd(ei, e, s, d);
    atomicAdd(&esum[d], __expf(leaky02(as[s] + ad[d]) - emax[d]));
}

__global__ void agg2_kernel(const int* __restrict__ ei, const float* __restrict__ h2,
                            const float* __restrict__ as, const float* __restrict__ ad,
                            const float* __restrict__ emax, const float* __restrict__ esum,
                            float* __restrict__ out2) {
    const int e = blockIdx.x * (blockDim.x >> 5) + (threadIdx.x >> 5);
    if (e >= EP) return;
    const int lane = threadIdx.x & 31;
    int s, d; edge_sd(ei, e, s, d);
    const float al = __expf(leaky02(as[s] + ad[d]) - emax[d]) / esum[d];
    const float* hp = h2 + (size_t)s * OUT_C;
    float* op = out2 + (size_t)d * OUT_C;
    atomicAdd(&op[lane],      hp[lane]      * al);
    atomicAdd(&op[lane + 32], hp[lane + 32] * al);
}

// ---------------------------------------------------------------------------
// final: out = log_softmax(out2 + b2) over 64 classes; one wave per node
// ---------------------------------------------------------------------------
__global__ void final_kernel(const float* __restrict__ out2, const float* __restrict__ b2,
                             float* __restrict__ out) {
    const int node = blockIdx.x * (blockDim.x >> 5) + (threadIdx.x >> 5);
    if (node >= N_NODES) return;
    const int lane = threadIdx.x & 31;
    const float* ip = out2 + (size_t)node * OUT_C;
    const float y0 = ip[lane] + b2[lane];
    const float y1 = ip[lane + 32] + b2[lane + 32];
    float m = fmaxf(y0, y1);
    #pragma unroll
    for (int off = 16; off > 0; off >>= 1) m = fmaxf(m, __shfl_xor(m, off, 32));
    float sx = __expf(y0 - m) + __expf(y1 - m);
    #pragma unroll
    for (int off = 16; off > 0; off >>= 1) sx += __shfl_xor(sx, off, 32);
    const float ls = __logf(sx);
    float* op = out + (size_t)node * OUT_C;
    op[lane]      = y0 - m - ls;
    op[lane + 32] = y1 - m - ls;
}

// ---------------------------------------------------------------------------
extern "C" void kernel_launch(void* const* d_in, const int* in_sizes, int n_in,
                              void* d_out, int out_size, void* d_ws, size_t ws_size,
                              hipStream_t stream) {
    const float* x     = (const float*)d_in[0];
    const int*   ei    = (const int*)d_in[1];
    const float* W1    = (const float*)d_in[2];
    const float* as1w  = (const float*)d_in[3];
    const float* ad1w  = (const float*)d_in[4];
    const float* b1    = (const float*)d_in[5];
    const float* W2    = (const float*)d_in[6];
    const float* as2w  = (const float*)d_in[7];
    const float* ad2w  = (const float*)d_in[8];
    const float* b2    = (const float*)d_in[9];
    float* out = (float*)d_out;

    float* ws    = (float*)d_ws;
    float* h1    = ws;                                   // N*512
    float* g     = h1 + (size_t)N_NODES * F1;            // N*512 (agg out -> elu -> layer2 in)
    float* h2    = g + (size_t)N_NODES * F1;             // N*64
    float* o2    = h2 + (size_t)N_NODES * OUT_C;         // N*64
    float* as1   = o2 + (size_t)N_NODES * OUT_C;         // N*8
    float* ad1   = as1 + (size_t)N_NODES * HEADS;
    float* emax1 = ad1 + (size_t)N_NODES * HEADS;
    float* esum1 = emax1 + (size_t)N_NODES * HEADS;
    float* as2   = esum1 + (size_t)N_NODES * HEADS;      // N
    float* ad2   = as2 + N_NODES;
    float* emax2 = ad2 + N_NODES;
    float* esum2 = emax2 + N_NODES;

    const int B = 256;
    const int waves_per_blk = B / 32;

    // layer 1 GEMM: h1 = x @ W1   (3125 * 32 tiles)
    {
        int tiles = (N_NODES / 16) * (F1 / 16);
        gemm_wmma_f32<<<(tiles + waves_per_blk - 1) / waves_per_blk, B, 0, stream>>>(
            x, W1, h1, N_NODES, IN_C, F1);
    }
    init_kernel<<<((size_t)N_NODES * F1 + B - 1) / B, B, 0, stream>>>(
        g, emax1, esum1, o2, emax2, esum2);
    alpha1_kernel<<<(N_NODES + waves_per_blk - 1) / waves_per_blk, B, 0, stream>>>(
        h1, as1w, ad1w, as1, ad1);
    max1_kernel<<<(EP * HEADS + B - 1) / B, B, 0, stream>>>(ei, as1, ad1, emax1);
    sum1_kernel<<<(EP * HEADS + B - 1) / B, B, 0, stream>>>(ei, as1, ad1, emax1, esum1);
    agg1_kernel<<<(EP + waves_per_blk - 1) / waves_per_blk, B, 0, stream>>>(
        ei, h1, as1, ad1, emax1, esum1, g);
    post1_kernel<<<((size_t)N_NODES * F1 + B - 1) / B, B, 0, stream>>>(g, b1);

    // layer 2 GEMM: h2 = g @ W2   (3125 * 4 tiles)
    {
        int tiles = (N_NODES / 16) * (OUT_C / 16);
        gemm_wmma_f32<<<(tiles + waves_per_blk - 1) / waves_per_blk, B, 0, stream>>>(
            g, W2, h2, N_NODES, F1, OUT_C);
    }
    alpha2_kernel<<<(N_NODES + waves_per_blk - 1) / waves_per_blk, B, 0, stream>>>(
        h2, as2w, ad2w, as2, ad2);
    max2_kernel<<<(EP + B - 1) / B, B, 0, stream>>>(ei, as2, ad2, emax2);
    sum2_kernel<<<(EP + B - 1) / B, B, 0, stream>>>(ei, as2, ad2, emax2, esum2);
    agg2_kernel<<<(EP + waves_per_blk - 1) / waves_per_blk, B, 0, stream>>>(
        ei, h2, as2, ad2, emax2, esum2, o2);
    final_kernel<<<(N_NODES + waves_per_blk - 1) / waves_per_blk, B, 0, stream>>>(o2, b2, out);
}